// CRFLoss_29910152249582
// MI455X (gfx1250) — compile-verified
//
#include <hip/hip_runtime.h>
#include <hip/hip_bf16.h>

#define SMALLV   (-1000.0f)
#define BATCH    512
#define SEQ      1024
#define NLAB     64
#define CST      66          // NLAB + 2 states
#define CPAD     68          // K padded to multiple of 4
#define NTILES   5           // ceil(66/16)
#define KCHUNKS  17          // 68 / 4
#define MB       16          // batches per workgroup tile
#define WAVES    5           // one wave per N-tile
#define TPB      (WAVES * 32)

typedef float v2f __attribute__((ext_vector_type(2)));
typedef float v8f __attribute__((ext_vector_type(8)));

// ---------------------------------------------------------------------------
// Forward (alpha) recursion in exp-domain with per-step rescaling.
//   new[j] = m_b + log( sum_i exp(prev[i]-m_b) * exp(T[i,j]) ) + ob[j]
// WMMA f32 16x16x4:  D(16x16) = A(16x4) x B(4x16) + C
//   A = exp(alpha - max) for 16 batches, B = exp(transitions) (constant).
// 5 waves per workgroup: wave w owns N-tile w (17 WMMAs/step on the critical
// path instead of 85). pred rows are double-buffered through LDS one step
// ahead (coalesced b128 loads) + global_prefetch two steps ahead.
// ---------------------------------------------------------------------------
__global__ __launch_bounds__(TPB, 1)
void crf_forward(const float* __restrict__ pred,
                 const int*   __restrict__ seq_len,
                 const float* __restrict__ trans,
                 float*       __restrict__ ws)
{
    __shared__ float Efrag[NTILES * KCHUNKS * 64];        // B fragments
    __shared__ float aexp [MB * CPAD];                    // exp(alpha - max)
    __shared__ float alpha[MB * CPAD];                    // log-domain state
    __shared__ __align__(16) float obsbuf[2][MB * NLAB];  // staged pred rows
    __shared__ float maxv [MB];
    __shared__ float pmax [32];
    __shared__ int   slen [MB];
    __shared__ int   tlim;

    const int tid   = threadIdx.x;
    const int lane  = tid & 31;
    const int wv    = tid >> 5;                  // wave index == N-tile
    const int bt    = blockIdx.x;                // batch tile
    const int mloc  = lane & 15;
    const int khalf = (lane >> 4) * 2;           // 0 or 2

    // --- Constant B fragments: E[k][n] = exp(T[k][n]); wave w builds tile w --
    {
        const int nt = wv;
        const int n  = nt * 16 + mloc;
        for (int kc = 0; kc < KCHUNKS; ++kc) {
            const int k0 = 4 * kc + khalf;
            float v0 = 0.f, v1 = 0.f;
            if (n < CST) {
                if (k0     < CST) v0 = __expf(trans[k0 * CST + n]);
                if (k0 + 1 < CST) v1 = __expf(trans[(k0 + 1) * CST + n]);
            }
            const int base = ((nt * KCHUNKS + kc) * 32 + lane) * 2;
            Efrag[base]     = v0;
            Efrag[base + 1] = v1;
        }
    }

    // --- Init alpha = b_s = {SMALL x64, 0, SMALL}; K-pads = -1e30 -----------
    for (int i = tid; i < MB * CPAD; i += TPB) {
        const int col = i % CPAD;
        float v;
        if      (col < NLAB)   v = SMALLV;
        else if (col == NLAB)  v = 0.0f;
        else if (col == CST-1) v = SMALLV;
        else                   v = -1.0e30f;
        alpha[i] = v;
    }
    if (tid < MB) slen[tid] = seq_len[bt * MB + tid];
    __syncthreads();
    if (tid == 0) {
        int m = 1;
        for (int i = 0; i < MB; ++i) m = (slen[i] > m) ? slen[i] : m;
        tlim = m + 1;
    }
    __syncthreads();
    const int Tlim = tlim;

    // --- Stage pred rows for step tt into obsbuf[tt&1] (coalesced b128) -----
    auto fill = [&](int tt) {
        const int buf = tt & 1;
        for (int i4 = tid; i4 < (MB * NLAB) / 4; i4 += TPB) {   // 256 float4
            const int m = i4 >> 4;
            const int c = (i4 & 15) * 4;
            const float4 v = *(const float4*)
                &pred[((size_t)(bt * MB + m) * SEQ + (tt - 1)) * NLAB + c];
            *(float4*)&obsbuf[buf][m * NLAB + c] = v;
        }
    };

    fill(1);                                      // prologue for t = 1
    __syncthreads();

    for (int t = 1; t <= Tlim; ++t) {
        // Stage t+1 into LDS; prefetch t+2 into cache (global_prefetch_b8)
        if (t + 1 <= SEQ && t + 1 <= Tlim) fill(t + 1);
        if (t + 2 <= SEQ && t + 2 <= Tlim && tid < 32) {
            const int row  = tid & 15;
            const int half = tid >> 4;
            __builtin_prefetch(
                &pred[((size_t)(bt * MB + row) * SEQ + (t + 1)) * NLAB + half * 32],
                0, 0);
        }

        // 1. per-batch max over 66 valid states (2 threads per row)
        if (tid < 32) {
            const int row = tid & 15;
            const int h   = tid >> 4;
            const float* rp = &alpha[row * CPAD];
            const int j0 = h * 33, j1 = j0 + 33;  // [0,33) and [33,66)
            float m = -3.0e38f;
            for (int j = j0; j < j1; ++j) m = fmaxf(m, rp[j]);
            pmax[tid] = m;
        }
        __syncthreads();
        if (tid < 16) maxv[tid] = fmaxf(pmax[tid], pmax[tid + 16]);
        __syncthreads();

        // 2. exp-domain rescale (pads -> 0)
        for (int i = tid; i < MB * CPAD; i += TPB) {
            const int row = i / CPAD;
            aexp[i] = __expf(alpha[i] - maxv[row]);
        }
        __syncthreads();

        // 3. wave wv: (16 x 68) x (68 x 16) via 17 WMMA f32 16x16x4
        v8f acc = {};
        for (int kc = 0; kc < KCHUNKS; ++kc) {
            const int k0 = 4 * kc + khalf;
            const v2f a = *(const v2f*)&aexp[mloc * CPAD + k0];
            const v2f b = *(const v2f*)&Efrag[((wv * KCHUNKS + kc) * 32 + lane) * 2];
            acc = __builtin_amdgcn_wmma_f32_16x16x4_f32(
                false, a, false, b, (short)0, acc, false, false);
        }

        // 4. back to log domain + observation, write new alpha (own tile)
        {
            const int n   = wv * 16 + mloc;
            const int buf = t & 1;
            if (n < CST) {
                #pragma unroll
                for (int r = 0; r < 8; ++r) {
                    const int m = r + 8 * (lane >> 4);
                    float obsv;
                    if (t <= SEQ)
                        obsv = (n < NLAB) ? obsbuf[buf][m * NLAB + n] : SMALLV;
                    else   // t == SEQ+1 : rp row {0 x64, SMALL, SMALL}
                        obsv = (n < NLAB) ? 0.0f : SMALLV;
                    if (t == slen[m] + 1)          // end marker e_s
                        obsv += (n < NLAB) ? SMALLV
                                           : ((n == NLAB) ? 0.0f : 1000.0f);
                    const float v = fmaxf(acc[r], 1.0e-30f);
                    alpha[m * CPAD + n] = __logf(v) + maxv[m] + obsv;
                }
            }
        }
        __syncthreads();

        // 5. capture per-batch final logsumexp at t == seq_len+1
        if (tid < 16 && t == slen[tid] + 1) {
            const float* row = &alpha[tid * CPAD];
            float m = -3.0e38f;
            for (int j = 0; j < CST; ++j) m = fmaxf(m, row[j]);
            float s = 0.f;
            for (int j = 0; j < CST; ++j) s += __expf(row[j] - m);
            ws[bt * MB + tid] = m + __logf(s);
        }
        __syncthreads();
    }
}

// ---------------------------------------------------------------------------
// Reference-path score: masked emission gather + transition chain.
// One block per batch; deterministic tree reduction -> ws2[b].
// ---------------------------------------------------------------------------
__global__ __launch_bounds__(256, 1)
void crf_realpath(const float* __restrict__ pred,
                  const int*   __restrict__ ref,
                  const int*   __restrict__ seq_len,
                  const float* __restrict__ trans,
                  float*       __restrict__ ws2)
{
    const int b   = blockIdx.x;
    const int tid = threadIdx.x;
    const int sl  = seq_len[b];

    float acc = 0.0f;
    for (int t = tid; t < sl; t += 256)
        acc += pred[((size_t)b * SEQ + t) * NLAB + ref[b * SEQ + t]];
    // tags: [64, ref[0..sl-1], 65] ; sum T[tag[j], tag[j+1]] for j = 0..sl
    for (int j = tid; j <= sl; j += 256) {
        const int from = (j == 0)  ? NLAB    : ref[b * SEQ + (j - 1)];
        const int to   = (j == sl) ? CST - 1 : ref[b * SEQ + j];
        acc += trans[from * CST + to];
    }

    __shared__ float red[256];
    red[tid] = acc;
    __syncthreads();
    for (int s = 128; s > 0; s >>= 1) {
        if (tid < s) red[tid] += red[tid + s];
        __syncthreads();
    }
    if (tid == 0) ws2[b] = red[0];
}

// ---------------------------------------------------------------------------
// Final deterministic combine: out = sum(ws) - sum(ws2)
// ---------------------------------------------------------------------------
__global__ __launch_bounds__(256, 1)
void crf_final(const float* __restrict__ ws,
               const float* __restrict__ ws2,
               float*       __restrict__ out)
{
    const int tid = threadIdx.x;
    float a = 0.0f;
    for (int i = tid; i < BATCH; i += 256) a += ws[i] - ws2[i];
    __shared__ float red[256];
    red[tid] = a;
    __syncthreads();
    for (int s = 128; s > 0; s >>= 1) {
        if (tid < s) red[tid] += red[tid + s];
        __syncthreads();
    }
    if (tid == 0) out[0] = red[0];
}

extern "C" void kernel_launch(void* const* d_in, const int* in_sizes, int n_in,
                              void* d_out, int out_size, void* d_ws, size_t ws_size,
                              hipStream_t stream)
{
    const float* pred    = (const float*)d_in[0];
    const int*   ref     = (const int*)  d_in[1];
    const int*   seq_len = (const int*)  d_in[2];
    const float* trans   = (const float*)d_in[3];

    float* ws  = (float*)d_ws;        // [BATCH] all-paths logsumexp per batch
    float* ws2 = ws + BATCH;          // [BATCH] real-path score per batch

    crf_forward <<<BATCH / MB, TPB, 0, stream>>>(pred, seq_len, trans, ws);
    crf_realpath<<<BATCH,      256, 0, stream>>>(pred, ref, seq_len, trans, ws2);
    crf_final   <<<1,          256, 0, stream>>>(ws, ws2, (float*)d_out);
}